// MahalanobisDistance_4904852652890
// MI455X (gfx1250) — compile-verified
//
#include <hip/hip_runtime.h>
#include <math.h>

typedef __attribute__((ext_vector_type(2))) float v2f;
typedef __attribute__((ext_vector_type(8))) float v8f;

#define BATCH 8
#define NPATCH 3136
#define DIM 256
#define PADM 16   // batch padded to 16 for WMMA M

// Reduce one 16x16 D-fragment (left = delta * cov_inv tile) against delta and
// atomically accumulate per-batch partial quadratic forms into s_acc[0..7].
__device__ __forceinline__ void reduce_tile(const v8f acc, int ctile, int lane,
                                            const float* s_delta, float* s_acc) {
    const int col   = ctile + (lane & 15);
    const int mbase = (lane < 16) ? 0 : 8;   // high half holds pad rows (zeros)
    float p[8];
#pragma unroll
    for (int i = 0; i < 8; ++i)
        p[i] = acc[i] * s_delta[col * PADM + mbase + i];
    // sum over the 16 columns held by this half-wave (xor 1,2,4,8 stays in half)
#pragma unroll
    for (int off = 1; off < 16; off <<= 1) {
#pragma unroll
        for (int i = 0; i < 8; ++i)
            p[i] += __shfl_xor(p[i], off, 32);
    }
    if (lane == 0) {
#pragma unroll
        for (int i = 0; i < 8; ++i)
            atomicAdd(&s_acc[i], p[i]);   // ds_add_f32; upper-half batches were exact zeros
    }
}

__global__ __launch_bounds__(256) void mahalanobis_wmma_kernel(
    const float* __restrict__ features,   // (B, N, D)
    const float* __restrict__ mean,       // (N, D)
    const float* __restrict__ cov_inv,    // (N, D, D)
    float* __restrict__ out)              // (B, N) == (B, 56, 56)
{
    __shared__ float s_delta[DIM * PADM]; // transposed: [d][m], 16 KB, rows m>=8 are zero
    __shared__ float s_acc[BATCH];

    const int n    = blockIdx.x;
    const int tid  = threadIdx.x;         // 0..255
    const int wave = tid >> 5;            // 0..7
    const int lane = tid & 31;

    // ---- stage delta = features - mean into LDS, transposed & zero-padded ----
    {
        const float mn = mean[(size_t)n * DIM + tid];
        float vals[BATCH];
#pragma unroll
        for (int b = 0; b < BATCH; ++b)
            vals[b] = features[(size_t)b * NPATCH * DIM + (size_t)n * DIM + tid] - mn;
#pragma unroll
        for (int b = 0; b < BATCH; ++b)
            s_delta[tid * PADM + b] = vals[b];
#pragma unroll
        for (int b = BATCH; b < PADM; ++b)
            s_delta[tid * PADM + b] = 0.0f;
    }
    if (tid < BATCH) s_acc[tid] = 0.0f;
    __syncthreads();

    // ---- 16x256x256 GEMM: left = deltaPad(16xD) * cov_inv[n](DxD) ----
    const float* covn = cov_inv + (size_t)n * DIM * DIM;
    const int m  = lane & 15;           // A: row M / B: col N within tile
    const int kq = (lane >> 4) * 2;     // K sub-pair selected by lane half

    const int c0 = wave * 32;           // this wave's 2 column tiles: c0, c0+16
    v8f acc0 = {};
    v8f acc1 = {};

#pragma unroll 4
    for (int k0 = 0; k0 < DIM; k0 += 4) {
        // A fragment (shared by both tiles): lane m holds delta[m][k0+kq(+1)]
        v2f a;
        a.x = s_delta[(k0 + kq) * PADM + m];
        a.y = s_delta[(k0 + kq + 1) * PADM + m];

        // B fragments: cov_inv rows k0+kq, k0+kq+1; streamed (no reuse -> NT)
        const float* r0 = covn + (size_t)(k0 + kq) * DIM;
        const float* r1 = r0 + DIM;
        v2f b0, b1;
        b0.x = __builtin_nontemporal_load(r0 + c0 + m);
        b0.y = __builtin_nontemporal_load(r1 + c0 + m);
        b1.x = __builtin_nontemporal_load(r0 + c0 + 16 + m);
        b1.y = __builtin_nontemporal_load(r1 + c0 + 16 + m);

        acc0 = __builtin_amdgcn_wmma_f32_16x16x4_f32(
            false, a, false, b0, (short)0, acc0, false, false);
        acc1 = __builtin_amdgcn_wmma_f32_16x16x4_f32(
            false, a, false, b1, (short)0, acc1, false, false);
    }

    // ---- dist2[b] += sum_col left[b][col] * delta[b][col] ----
    reduce_tile(acc0, c0,      lane, s_delta, s_acc);
    reduce_tile(acc1, c0 + 16, lane, s_delta, s_acc);

    __syncthreads();
    if (tid < BATCH) {
        float d2 = s_acc[tid];
        out[(size_t)tid * NPATCH + n] = sqrtf(fmaxf(d2, 0.0f));
    }
}

extern "C" void kernel_launch(void* const* d_in, const int* in_sizes, int n_in,
                              void* d_out, int out_size, void* d_ws, size_t ws_size,
                              hipStream_t stream) {
    const float* features = (const float*)d_in[0];  // (8, 3136, 256) f32
    const float* mean     = (const float*)d_in[1];  // (3136, 256) f32
    const float* cov_inv  = (const float*)d_in[2];  // (3136, 256, 256) f32
    // d_in[3] = width, d_in[4] = height (unused: fixed 56x56)
    float* out = (float*)d_out;                     // (8, 56, 56) f32

    mahalanobis_wmma_kernel<<<NPATCH, 256, 0, stream>>>(features, mean, cov_inv, out);
}